// Encoder_6966436954610
// MI455X (gfx1250) — compile-verified
//
#include <hip/hip_runtime.h>
#include <math.h>

typedef float v2f __attribute__((ext_vector_type(2)));
typedef float v8f __attribute__((ext_vector_type(8)));

// ---------------------------------------------------------------------------
// Problem: B=32, N=20 qubits. out[b, k] (complex64) = prod_q v[b,q,bit_q(k)]
// where bit_q(k) = (k >> (19-q)) & 1 (qubit 0 = MSB),
//   v[b,q,0] = cos(ry/2) * e^{-i rz/2},  v[b,q,1] = sin(ry/2) * e^{+i rz/2}.
// Factorize: k = i*1024 + j  ->  amp = hi[b,i] * lo[b,j]  (complex multiply).
// hi covers qubits 0..9, lo covers qubits 10..19.
// ---------------------------------------------------------------------------

#define NB   32
#define NQ   20
#define HALF 1024   // 2^10 partial amplitudes per half

// ---------- Kernel 1: per-batch partial amplitudes (512 KB into d_ws) -------
__global__ void qsim_partials(const float* __restrict__ ry,
                              const float* __restrict__ rz,
                              float2* __restrict__ hi_part,   // [NB][HALF]
                              float2* __restrict__ lo_part) { // [NB][HALF]
    __shared__ float s_cos[NQ], s_sin[NQ], s_hz[NQ];
    const int b = blockIdx.x;
    const int t = threadIdx.x;
    if (t < NQ) {
        float hry = 0.5f * ry[b * NQ + t];
        float s, c;
        __sincosf(hry, &s, &c);
        s_cos[t] = c;
        s_sin[t] = s;
        s_hz[t]  = 0.5f * rz[b * NQ + t];
    }
    __syncthreads();

    // 256 threads x 4 indices each = 1024 entries per half
    for (int rep = 0; rep < 4; ++rep) {
        const int j = t + rep * 256;
        // ---- high half: qubits 0..9, bit_q = (j >> (9-q)) & 1
        float m = 1.0f, p = 0.0f;
        #pragma unroll
        for (int q = 0; q < 10; ++q) {
            const int bit = (j >> (9 - q)) & 1;
            m *= bit ? s_sin[q] : s_cos[q];
            p += bit ? s_hz[q] : -s_hz[q];
        }
        float sp, cp;
        __sincosf(p, &sp, &cp);
        hi_part[b * HALF + j] = make_float2(m * cp, m * sp);

        // ---- low half: qubits 10..19, bit_q = (j >> (9-qq)) & 1
        m = 1.0f; p = 0.0f;
        #pragma unroll
        for (int qq = 0; qq < 10; ++qq) {
            const int bit = (j >> (9 - qq)) & 1;
            m *= bit ? s_sin[10 + qq] : s_cos[10 + qq];
            p += bit ? s_hz[10 + qq] : -s_hz[10 + qq];
        }
        __sincosf(p, &sp, &cp);
        lo_part[b * HALF + j] = make_float2(m * cp, m * sp);
    }
}

// ---------- Kernel 2: complex outer product via V_WMMA_F32_16X16X4_F32 ------
// One wave computes one 16x16 complex tile: D_re = hr*lr^T - hi*li^T,
// D_im = hr*li^T + hi*lr^T, using K=2 of the K=4 f32 WMMA (K=2,3 zeroed).
//
// f32 16x4 A layout: lanes 0-15 (M=lane) hold K=0 (VGPR0) and K=1 (VGPR1);
// lanes 16-31 hold K=2,3 -> zero there. B 4x16 mirrors this (N=lane%16).
__global__ void __launch_bounds__(256)
qsim_outer(const float2* __restrict__ hi_part,
           const float2* __restrict__ lo_part,
           float2* __restrict__ out) {
    const int wave   = (blockIdx.x * blockDim.x + threadIdx.x) >> 5;
    const int lane   = threadIdx.x & 31;
    const int b      = wave >> 12;            // 4096 tiles per batch
    const int tile   = wave & 4095;
    const int tileHi = tile >> 6;             // 64 tiles along hi dim
    const int tileLo = tile & 63;             // 64 tiles along lo dim
    const int n16    = lane & 15;
    const int upper  = lane >> 4;             // lanes 16..31 -> K=2,3 (zeros)

    const float2 a = hi_part[b * HALF + tileHi * 16 + n16]; // row factor (M)
    const float2 l = lo_part[b * HALF + tileLo * 16 + n16]; // col factor (N)

    const float z = upper ? 0.0f : 1.0f;      // zero K=2,3 contribution
    v2f A, Bre, Bim;
    A.x   = z * a.x;  A.y   = z * a.y;        // [hr, hi]
    Bre.x = z * l.x;  Bre.y = z * -l.y;       // [lr, -li]
    Bim.x = z * l.y;  Bim.y = z * l.x;        // [li,  lr]

    v8f c = {};
    // D = A x B + 0 : two K=4 f32 WMMAs per 256 complex outputs
    v8f dre = __builtin_amdgcn_wmma_f32_16x16x4_f32(
        false, A, false, Bre, (short)0, c, false, false);
    v8f dim = __builtin_amdgcn_wmma_f32_16x16x4_f32(
        false, A, false, Bim, (short)0, c, false, false);

    // C/D layout: VGPR r -> M = r (lanes 0-15) or r+8 (lanes 16-31); N = lane%16
    const size_t base = (size_t)b * (HALF * HALF)
                      + (size_t)(tileHi * 16) * HALF
                      + (size_t)(tileLo * 16);
    #pragma unroll
    for (int r = 0; r < 8; ++r) {
        const size_t idx = base + (size_t)(r + 8 * upper) * HALF + n16;
        v2f val; val.x = dre[r]; val.y = dim[r];
        __builtin_nontemporal_store(val, (v2f*)&out[idx]);  // streaming: 256MB > L2
    }
}

// ---------------------------------------------------------------------------
extern "C" void kernel_launch(void* const* d_in, const int* in_sizes, int n_in,
                              void* d_out, int out_size, void* d_ws, size_t ws_size,
                              hipStream_t stream) {
    const float* ry = (const float*)d_in[0];   // [32,20] float32
    const float* rz = (const float*)d_in[1];   // [32,20] float32
    float2* out = (float2*)d_out;              // [32, 2^20] complex64 (re,im pairs)

    // workspace: hi_part [32][1024] float2, lo_part [32][1024] float2 = 512 KB
    float2* hi_part = (float2*)d_ws;
    float2* lo_part = hi_part + NB * HALF;

    qsim_partials<<<NB, 256, 0, stream>>>(ry, rz, hi_part, lo_part);

    // 32 batches * 4096 tiles = 131072 waves; 8 waves (256 thr) per block
    qsim_outer<<<(NB * 4096) / 8, 256, 0, stream>>>(hi_part, lo_part, out);
}